// DfMap_15109694947350
// MI455X (gfx1250) — compile-verified
//
#include <hip/hip_runtime.h>

#define H 256
#define W 256
#define HW 65536
#define BS 8
#define CIN 16
#define COUT 32
#define NSTEPS 7
#define PAIRS 128            // BS*CIN flow fields
#define BN_EPS 1e-5f
#define INV_CNT (1.0f / 524288.0f)   // 1/(BS*H*W)
#define VSCALE (1.0f / 128.0f)       // 1/2^NSTEPS

typedef __attribute__((ext_vector_type(2))) float v2f;
typedef __attribute__((ext_vector_type(8))) float v8f;

// ---------------------------------------------------------------- utilities

__device__ __forceinline__ float bsample1(const float* __restrict__ p,
                                          float py, float px) {
    float yf = floorf(py), xf = floorf(px);
    int y0 = (int)yf, x0 = (int)xf;
    int y1 = y0 + 1, x1 = x0 + 1;
    float wy1 = py - yf, wx1 = px - xf;
    float wy0 = 1.0f - wy1, wx0 = 1.0f - wx1;
    float my0 = (y0 >= 0 && y0 < H) ? 1.0f : 0.0f;
    float my1 = (y1 >= 0 && y1 < H) ? 1.0f : 0.0f;
    float mx0 = (x0 >= 0 && x0 < W) ? 1.0f : 0.0f;
    float mx1 = (x1 >= 0 && x1 < W) ? 1.0f : 0.0f;
    int yc0 = min(max(y0, 0), H - 1), yc1 = min(max(y1, 0), H - 1);
    int xc0 = min(max(x0, 0), W - 1), xc1 = min(max(x1, 0), W - 1);
    float w00 = wy0 * wx0 * my0 * mx0, w01 = wy0 * wx1 * my0 * mx1;
    float w10 = wy1 * wx0 * my1 * mx0, w11 = wy1 * wx1 * my1 * mx1;
    return w00 * p[yc0 * W + xc0] + w01 * p[yc0 * W + xc1] +
           w10 * p[yc1 * W + xc0] + w11 * p[yc1 * W + xc1];
}

__device__ __forceinline__ void bsample2(const float* __restrict__ p0,
                                         const float* __restrict__ p1,
                                         float py, float px,
                                         float& r0, float& r1) {
    float yf = floorf(py), xf = floorf(px);
    int y0 = (int)yf, x0 = (int)xf;
    int y1 = y0 + 1, x1 = x0 + 1;
    float wy1 = py - yf, wx1 = px - xf;
    float wy0 = 1.0f - wy1, wx0 = 1.0f - wx1;
    float my0 = (y0 >= 0 && y0 < H) ? 1.0f : 0.0f;
    float my1 = (y1 >= 0 && y1 < H) ? 1.0f : 0.0f;
    float mx0 = (x0 >= 0 && x0 < W) ? 1.0f : 0.0f;
    float mx1 = (x1 >= 0 && x1 < W) ? 1.0f : 0.0f;
    int yc0 = min(max(y0, 0), H - 1), yc1 = min(max(y1, 0), H - 1);
    int xc0 = min(max(x0, 0), W - 1), xc1 = min(max(x1, 0), W - 1);
    int i00 = yc0 * W + xc0, i01 = yc0 * W + xc1;
    int i10 = yc1 * W + xc0, i11 = yc1 * W + xc1;
    float w00 = wy0 * wx0 * my0 * mx0, w01 = wy0 * wx1 * my0 * mx1;
    float w10 = wy1 * wx0 * my1 * mx0, w11 = wy1 * wx1 * my1 * mx1;
    r0 = w00 * p0[i00] + w01 * p0[i01] + w10 * p0[i10] + w11 * p0[i11];
    r1 = w00 * p1[i00] + w01 * p1[i01] + w10 * p1[i10] + w11 * p1[i11];
}

// ---------------------------------------------------------------- kernels

__global__ void zero_stats_kernel(float* __restrict__ stats) {
    if (threadIdx.x < 64) stats[threadIdx.x] = 0.0f;
}

__global__ void init_out_kernel(float* __restrict__ out,
                                const float* __restrict__ fuse_b) {
    int i = blockIdx.x * 256 + threadIdx.x;      // BS*CIN*HW elements
    out[i] = fuse_b[(i >> 16) & (CIN - 1)];
}

// 3x3 conv (pad 1) + bias; per-block partial sums of x and x^2 for BN stats.
__global__ __launch_bounds__(256) void conv_bn_stats_kernel(
        const float* __restrict__ f, const float* __restrict__ vw,
        const float* __restrict__ vb, float* __restrict__ vecraw,
        float* __restrict__ stats) {
    __shared__ float wlds[CIN * 9];
    __shared__ float red[256];
    const int tid = threadIdx.x;
    const int oc = blockIdx.y, b = blockIdx.z;
    if (tid < CIN * 9) wlds[tid] = vw[oc * CIN * 9 + tid];
    __syncthreads();

    const int pix = blockIdx.x * 256 + tid;
    const int y = pix >> 8, x = pix & (W - 1);
    float acc = vb[oc];
    #pragma unroll
    for (int ic = 0; ic < CIN; ++ic) {
        const float* fp = f + ((size_t)(b * CIN + ic) << 16);
        #pragma unroll
        for (int ky = 0; ky < 3; ++ky) {
            int yy = y + ky - 1;
            if (yy < 0 || yy >= H) continue;
            const float* rowp = fp + yy * W;
            #pragma unroll
            for (int kx = 0; kx < 3; ++kx) {
                int xx = x + kx - 1;
                if (xx < 0 || xx >= W) continue;
                acc += rowp[xx] * wlds[ic * 9 + ky * 3 + kx];
            }
        }
    }
    vecraw[((size_t)(b * COUT + oc) << 16) + pix] = acc;

    red[tid] = acc;
    __syncthreads();
    for (int s = 128; s > 0; s >>= 1) {
        if (tid < s) red[tid] += red[tid + s];
        __syncthreads();
    }
    if (tid == 0) atomicAdd(&stats[oc], red[0]);
    __syncthreads();
    red[tid] = acc * acc;
    __syncthreads();
    for (int s = 128; s > 0; s >>= 1) {
        if (tid < s) red[tid] += red[tid + s];
        __syncthreads();
    }
    if (tid == 0) atomicAdd(&stats[COUT + oc], red[0]);
}

// Per-channel affine: a = gamma*rsqrt(var+eps)/128, b = (beta - mean*..)/128.
__global__ void bn_finalize_kernel(const float* __restrict__ gamma,
                                   const float* __restrict__ beta,
                                   float* __restrict__ stats) {
    int c = threadIdx.x;
    if (c < COUT) {
        float mean = stats[c] * INV_CNT;
        float var = stats[COUT + c] * INV_CNT - mean * mean;
        float inv = rsqrtf(var + BN_EPS);
        float sc = gamma[c] * inv;
        stats[64 + c] = sc * VSCALE;
        stats[96 + c] = (beta[c] - mean * sc) * VSCALE;
    }
}

// In-place normalize + 1/2^NSTEPS scale; vecraw becomes flow field 0.
__global__ __launch_bounds__(256) void bn_apply_kernel(
        float* __restrict__ vec, const float* __restrict__ stats) {
    int i = blockIdx.x * 256 + threadIdx.x;     // BS*COUT*HW elements
    int ch = (i >> 16) & (COUT - 1);
    vec[i] = vec[i] * stats[64 + ch] + stats[96 + ch];
}

// One scaling-and-squaring step: out = in + warp(in, in).
__global__ __launch_bounds__(256) void vecint_step_kernel(
        const float* __restrict__ fin, float* __restrict__ fout) {
    int g = blockIdx.x * 256 + threadIdx.x;     // PAIRS*HW threads
    int p = g >> 16;
    int pix = g & (HW - 1);
    int y = pix >> 8, x = pix & (W - 1);
    const float* p0 = fin + (size_t)p * 2 * HW;
    const float* p1 = p0 + HW;
    float dy = p0[pix], dx = p1[pix];
    float s0, s1;
    bsample2(p0, p1, (float)y + dy, (float)x + dx, s0, s1);
    fout[(size_t)p * 2 * HW + pix]      = dy + s0;
    fout[(size_t)p * 2 * HW + HW + pix] = dx + s1;
}

// Warp f by the current deform and fold into out with W_s via f32 WMMA:
// per wave: Out(16 o x 16 px) += W_s(16x16) x M(16 c x 16 px), K split 4x4.
__global__ __launch_bounds__(256) void fuse_wmma_kernel(
        const float* __restrict__ f, const float* __restrict__ deform,
        const float* __restrict__ fuse_w, int step, float* __restrict__ out) {
    __shared__ float wlds[256];                 // wlds[o*16+c] = W_s[o][c]
    const int t = threadIdx.x;
    wlds[t] = fuse_w[(t >> 4) * (CIN * NSTEPS) + (t & 15) * NSTEPS + step];
    __syncthreads();

    const int wave = t >> 5, lane = t & 31;
    const int gw = blockIdx.x * 8 + wave;       // 32768 waves total
    const int b = gw >> 12;                     // 4096 pixel tiles per batch
    const int tile = gw & 4095;
    const int pixbase = tile << 4;
    const int half = lane >> 4, l = lane & 15;
    const int pix = pixbase + l;
    const int y = pix >> 8, x = pix & (W - 1);

    // B fragments: lane covers channels {4j + 2*half, +1}, j = 0..3
    v2f bm[4];
    #pragma unroll
    for (int j = 0; j < 4; ++j) {
        int c0 = 4 * j + 2 * half;
        const float* d0 = deform + (size_t)(b * CIN + c0) * 2 * HW;
        float m0 = bsample1(f + ((size_t)(b * CIN + c0) << 16),
                            (float)y + d0[pix], (float)x + d0[HW + pix]);
        const float* d1 = deform + (size_t)(b * CIN + c0 + 1) * 2 * HW;
        float m1 = bsample1(f + ((size_t)(b * CIN + c0 + 1) << 16),
                            (float)y + d1[pix], (float)x + d1[HW + pix]);
        bm[j].x = m0;
        bm[j].y = m1;
    }
    // A fragments: lane l holds row M=l; K pair picked by lane half
    v2f am[4];
    #pragma unroll
    for (int j = 0; j < 4; ++j) {
        int k = 4 * j + 2 * half;
        am[j].x = wlds[l * 16 + k];
        am[j].y = wlds[l * 16 + k + 1];
    }
    // C/D tile: VGPR r -> row (r + 8*half), col (pixbase + l)
    float* ob = out + ((size_t)(b * CIN) << 16);
    const int orow = 8 * half;
    union { v8f v; float s[8]; } c;
    #pragma unroll
    for (int r = 0; r < 8; ++r)
        c.s[r] = ob[(size_t)(orow + r) * HW + pixbase + l];
    #pragma unroll
    for (int j = 0; j < 4; ++j)
        c.v = __builtin_amdgcn_wmma_f32_16x16x4_f32(
            false, am[j], false, bm[j], (short)0, c.v, false, false);
    #pragma unroll
    for (int r = 0; r < 8; ++r)
        ob[(size_t)(orow + r) * HW + pixbase + l] = c.s[r];
}

// ---------------------------------------------------------------- launcher

extern "C" void kernel_launch(void* const* d_in, const int* in_sizes, int n_in,
                              void* d_out, int out_size, void* d_ws, size_t ws_size,
                              hipStream_t stream) {
    const float* f        = (const float*)d_in[0];
    const float* vec_w    = (const float*)d_in[1];
    const float* vec_b    = (const float*)d_in[2];
    const float* bn_gamma = (const float*)d_in[3];
    const float* bn_beta  = (const float*)d_in[4];
    const float* fuse_w   = (const float*)d_in[5];
    const float* fuse_b   = (const float*)d_in[6];
    float* out = (float*)d_out;

    const size_t VEC = (size_t)BS * COUT * HW;      // 16,777,216 floats
    float* vecraw = (float*)d_ws;                   // doubles as flow buffer A
    float* flowB  = vecraw + VEC;                   // flow buffer B
    float* stats  = flowB + VEC;                    // 128 floats

    zero_stats_kernel<<<1, 64, 0, stream>>>(stats);
    init_out_kernel<<<(BS * CIN * HW) / 256, 256, 0, stream>>>(out, fuse_b);

    dim3 cgrid(HW / 256, COUT, BS);
    conv_bn_stats_kernel<<<cgrid, 256, 0, stream>>>(f, vec_w, vec_b, vecraw, stats);
    bn_finalize_kernel<<<1, 32, 0, stream>>>(bn_gamma, bn_beta, stats);
    bn_apply_kernel<<<VEC / 256, 256, 0, stream>>>(vecraw, stats);

    float* cur = vecraw;
    float* nxt = flowB;
    for (int s = 0; s < NSTEPS; ++s) {
        vecint_step_kernel<<<((size_t)PAIRS * HW) / 256, 256, 0, stream>>>(cur, nxt);
        fuse_wmma_kernel<<<(BS * (HW / 16)) / 8, 256, 0, stream>>>(f, nxt, fuse_w, s, out);
        float* tmp = cur; cur = nxt; nxt = tmp;
    }
}